// SAFNN_84851373899961
// MI455X (gfx1250) — compile-verified
//
#include <hip/hip_runtime.h>

typedef float v2f __attribute__((ext_vector_type(2)));
typedef float v8f __attribute__((ext_vector_type(8)));

#define B_SZ   4096
#define F_NUM  16
#define V_SZ   20000
#define E_SZ   64
#define HID    1024

// ---------------------------------------------------------------------------
// Kernel 1: embedding gather.  xw[b, f*64+e] = emb[f, idx[b,f], e]
// ---------------------------------------------------------------------------
__global__ void __launch_bounds__(256) gather_embed(const int* __restrict__ idx,
                                                    const float* __restrict__ emb,
                                                    float* __restrict__ xw) {
    int tid = blockIdx.x * 256 + threadIdx.x;   // 0 .. B*HID-1
    int b = tid >> 10;
    int c = tid & 1023;
    int f = c >> 6;
    int e = c & 63;
    int v = idx[b * F_NUM + f];
    xw[tid] = emb[((size_t)f * V_SZ + (size_t)v) * E_SZ + e];
}

// ---------------------------------------------------------------------------
// Kernel 2: Y = relu(X @ W + bias), X:[4096,1024], W:[1024,1024] row-major.
// One wave -> 16x64 output tile via V_WMMA_F32_16X16X4_F32.
// Fragment layouts (CDNA5 ISA 7.12.2, f32):
//   A (16x4):  elem(vgpr j, lane l) = A[l%16, k0 + 2*(l>>4) + j]   (2 VGPRs)
//   B (4x16):  elem(vgpr j, lane l) = B[k0 + 2*(l>>4) + j, l%16]   (2 VGPRs)
//   C (16x16): elem(vgpr j, lane l) = C[8*(l>>4) + j, l%16]        (8 VGPRs)
// ---------------------------------------------------------------------------
__global__ void __launch_bounds__(128) gemm_relu(const float* __restrict__ X,
                                                 const float* __restrict__ W,
                                                 const float* __restrict__ bias,
                                                 float* __restrict__ Y) {
    const int lane = threadIdx.x & 31;
    const int gw   = blockIdx.x * 4 + (threadIdx.x >> 5);  // global wave id
    const int m0   = (gw >> 4) * 16;                        // 256 M-tiles
    const int n0   = (gw & 15) * 64;                        // 16 N-chunks of 64
    const int lmod = lane & 15;
    const int lhi  = lane >> 4;                             // 0 or 1

    v8f acc[4] = {v8f{}, v8f{}, v8f{}, v8f{}};

    const float* xrow = X + (size_t)(m0 + lmod) * HID + 2 * lhi;
    const float* wcol = W + (size_t)(2 * lhi) * HID + n0 + lmod;

    for (int k = 0; k < HID; k += 4) {
        // A fragment: two consecutive K values -> single 8B load
        v2f a = *(const v2f*)(xrow + k);
        const float* wr0 = wcol + (size_t)k * HID;  // row k + 2*lhi
        const float* wr1 = wr0 + HID;               // row k + 2*lhi + 1
        v2f b0 = { wr0[ 0], wr1[ 0] };
        v2f b1 = { wr0[16], wr1[16] };
        v2f b2 = { wr0[32], wr1[32] };
        v2f b3 = { wr0[48], wr1[48] };
        acc[0] = __builtin_amdgcn_wmma_f32_16x16x4_f32(false, a, false, b0, (short)0, acc[0], false, false);
        acc[1] = __builtin_amdgcn_wmma_f32_16x16x4_f32(false, a, false, b1, (short)0, acc[1], false, false);
        acc[2] = __builtin_amdgcn_wmma_f32_16x16x4_f32(false, a, false, b2, (short)0, acc[2], false, false);
        acc[3] = __builtin_amdgcn_wmma_f32_16x16x4_f32(false, a, false, b3, (short)0, acc[3], false, false);
    }

    // Store with bias + ReLU.  elem(j,l) of acc[s] -> Y[m0+8*lhi+j, n0+16*s+lmod]
    #pragma unroll
    for (int s = 0; s < 4; ++s) {
        const int n  = n0 + 16 * s + lmod;
        const float bv = bias[n];
        float* yp = Y + (size_t)(m0 + 8 * lhi) * HID + n;
        #pragma unroll
        for (int j = 0; j < 8; ++j) {
            float v = acc[s][j] + bv;
            yp[(size_t)j * HID] = v > 0.f ? v : 0.f;
        }
    }
}

// ---------------------------------------------------------------------------
// Kernel 3: layer-attention + final logit, one 128-thread block per batch row.
// Faithful to the reference's concat(axis=0).reshape(B,3,HID):
//   H[b,j] = cat[3*b + j], cat = [l0; l1; l2]  (row r: layer r/B, row r%B)
// S has a *zero* diagonal (masked before softmax, so diag contributes exp(0)).
// ---------------------------------------------------------------------------
__global__ void __launch_bounds__(128) attn_final(const float* __restrict__ L0,
                                                  const float* __restrict__ L1,
                                                  const float* __restrict__ L2,
                                                  const float* __restrict__ LF,
                                                  const float* __restrict__ wf,
                                                  const float* __restrict__ bf,
                                                  float* __restrict__ out) {
    const int b = blockIdx.x;
    const int t = threadIdx.x;
    const float* Ls[3] = {L0, L1, L2};
    const float* H[3];
    #pragma unroll
    for (int j = 0; j < 3; ++j) {
        int r = 3 * b + j;
        H[j] = Ls[r / B_SZ] + (size_t)(r % B_SZ) * HID;
    }

    __shared__ float red[3][128];
    __shared__ float cs[3];

    float p01 = 0.f, p02 = 0.f, p12 = 0.f;
    for (int h = t; h < HID; h += 128) {
        float h0 = H[0][h], h1 = H[1][h], h2 = H[2][h];
        p01 += h0 * h1; p02 += h0 * h2; p12 += h1 * h2;
    }
    red[0][t] = p01; red[1][t] = p02; red[2][t] = p12;
    __syncthreads();
    for (int off = 64; off > 0; off >>= 1) {
        if (t < off) {
            red[0][t] += red[0][t + off];
            red[1][t] += red[1][t + off];
            red[2][t] += red[2][t + off];
        }
        __syncthreads();
    }

    if (t == 0) {
        float s01 = red[0][0], s02 = red[1][0], s12 = red[2][0];
        float S[3][3] = {{0.f, s01, s02}, {s01, 0.f, s12}, {s02, s12, 0.f}};
        float col[3] = {0.f, 0.f, 0.f};
        #pragma unroll
        for (int i = 0; i < 3; ++i) {
            float m = fmaxf(S[i][0], fmaxf(S[i][1], S[i][2]));
            float e0 = expf(S[i][0] - m), e1 = expf(S[i][1] - m), e2 = expf(S[i][2] - m);
            float inv = 1.f / (e0 + e1 + e2);
            col[0] += e0 * inv; col[1] += e1 * inv; col[2] += e2 * inv;
        }
        cs[0] = col[0]; cs[1] = col[1]; cs[2] = col[2];
    }
    __syncthreads();

    const float c0 = cs[0], c1 = cs[1], c2 = cs[2];
    const float* lf = LF + (size_t)b * HID;
    float p = 0.f;
    for (int h = t; h < HID; h += 128) {
        float G = c0 * H[0][h] + c1 * H[1][h] + c2 * H[2][h];
        p += lf[h] * wf[h] + G * wf[HID + h];
    }
    __syncthreads();
    red[0][t] = p;
    __syncthreads();
    for (int off = 64; off > 0; off >>= 1) {
        if (t < off) red[0][t] += red[0][t + off];
        __syncthreads();
    }
    if (t == 0) out[b] = red[0][0] + bf[0];
}

// ---------------------------------------------------------------------------
// Launch: gather -> 4x gemm_relu (WMMA f32) -> attention+final
// Workspace layout (64 MB): ws0 = xw then l2, ws1 = l0, ws2 = l1, ws3 = l_final
// ---------------------------------------------------------------------------
extern "C" void kernel_launch(void* const* d_in, const int* in_sizes, int n_in,
                              void* d_out, int out_size, void* d_ws, size_t ws_size,
                              hipStream_t stream) {
    const int*   idx = (const int*)  d_in[0];
    const float* emb = (const float*)d_in[1];
    const float* w0  = (const float*)d_in[2];
    const float* b0  = (const float*)d_in[3];
    const float* w1  = (const float*)d_in[4];
    const float* b1  = (const float*)d_in[5];
    const float* w2  = (const float*)d_in[6];
    const float* b2  = (const float*)d_in[7];
    const float* w3  = (const float*)d_in[8];
    const float* b3  = (const float*)d_in[9];
    const float* wf  = (const float*)d_in[10];
    const float* bfin= (const float*)d_in[11];
    float* out = (float*)d_out;

    const size_t buf = (size_t)B_SZ * HID;  // 4M floats = 16 MB
    float* ws0 = (float*)d_ws;              // xw, later l2
    float* ws1 = ws0 + buf;                 // l0
    float* ws2 = ws1 + buf;                 // l1
    float* ws3 = ws2 + buf;                 // l_final

    // 1) gather: B*HID elements
    gather_embed<<<(B_SZ * HID) / 256, 256, 0, stream>>>(idx, emb, ws0);

    // 2) GEMM chain: 256 M-tiles * 16 N-chunks = 4096 waves, 4 waves/block
    const int gemm_blocks = 1024;
    gemm_relu<<<gemm_blocks, 128, 0, stream>>>(ws0, w0, b0, ws1);  // l0
    gemm_relu<<<gemm_blocks, 128, 0, stream>>>(ws1, w1, b1, ws2);  // l1
    gemm_relu<<<gemm_blocks, 128, 0, stream>>>(ws2, w2, b2, ws0);  // l2 (xw dead)
    gemm_relu<<<gemm_blocks, 128, 0, stream>>>(ws0, w3, b3, ws3);  // l_final

    // 3) attention + final logit
    attn_final<<<B_SZ, 128, 0, stream>>>(ws1, ws2, ws0, ws3, wf, bfin, out);
}